// SparseNeighborhoodAggregation_16630113370616
// MI455X (gfx1250) — compile-verified
//
#include <hip/hip_runtime.h>

typedef __attribute__((ext_vector_type(4))) float v4f;

// ---------------------------------------------------------------------------
// Kernel 1: zero the output (harness poisons d_out before timing).
// ---------------------------------------------------------------------------
__global__ void sna_zero_kernel(float* __restrict__ out, int n) {
  int i = blockIdx.x * blockDim.x + threadIdx.x;
  int n4 = n >> 2;
  if (i < n4) {
    v4f z = {0.0f, 0.0f, 0.0f, 0.0f};
    ((v4f*)out)[i] = z;
  }
  // scalar tail (n not multiple of 4): first (n & 3) threads handle it
  int tail = n - (n4 << 2);
  if (i < tail) out[(n4 << 2) + i] = 0.0f;
}

// ---------------------------------------------------------------------------
// Kernel 2: scatter-add. CH = F/4 float4-chunks per edge row; CH consecutive
// lanes cooperate on one edge row -> coalesced global_load_b128 (NT hint),
// then 4x native global_atomic_add_f32 at device scope (resolve in L2).
// ---------------------------------------------------------------------------
template <int CH>
__global__ void sna_scatter_kernel(const int* __restrict__ src,
                                   const float* __restrict__ w,
                                   float* __restrict__ out,
                                   unsigned int total) {
  unsigned int i = blockIdx.x * blockDim.x + threadIdx.x;
  if (i >= total) return;

  unsigned int e = i / CH;          // compile-time divisor -> mul/shift
  unsigned int c = i - e * CH;

  int s = src[e];

  const v4f* row = (const v4f*)(w + (size_t)e * (CH * 4));
  v4f v = __builtin_nontemporal_load(row + c);   // streaming, use-once

  float* dst = out + (size_t)s * (CH * 4) + (size_t)c * 4;

  // Native CDNA5 f32 atomic adds, no-return (STOREcnt), device scope so the
  // read-modify-write happens at L2 (WGP$ is read-only on CDNA5).
  asm volatile(
      "global_atomic_add_f32 %0, %1, off scope:SCOPE_DEV\n\t"
      "global_atomic_add_f32 %0, %2, off offset:4 scope:SCOPE_DEV\n\t"
      "global_atomic_add_f32 %0, %3, off offset:8 scope:SCOPE_DEV\n\t"
      "global_atomic_add_f32 %0, %4, off offset:12 scope:SCOPE_DEV"
      :
      : "v"(dst), "v"(v.x), "v"(v.y), "v"(v.z), "v"(v.w)
      : "memory");
}

// Generic fallback for F not divisible by 4 / unexpected shapes:
// one thread per (edge, feature) scalar element.
__global__ void sna_scatter_generic(const int* __restrict__ src,
                                    const float* __restrict__ w,
                                    float* __restrict__ out,
                                    long long total, int F) {
  long long i = (long long)blockIdx.x * blockDim.x + threadIdx.x;
  if (i >= total) return;
  long long e = i / F;
  int f = (int)(i - e * (long long)F);
  float val = w[i];
  float* dst = out + (size_t)src[e] * F + f;
  asm volatile("global_atomic_add_f32 %0, %1, off scope:SCOPE_DEV"
               :
               : "v"(dst), "v"(val)
               : "memory");
}

// ---------------------------------------------------------------------------
// Launch
// ---------------------------------------------------------------------------
extern "C" void kernel_launch(void* const* d_in, const int* in_sizes, int n_in,
                              void* d_out, int out_size, void* d_ws, size_t ws_size,
                              hipStream_t stream) {
  const int*   edge   = (const int*)d_in[0];    // [2, E] int32, row 0 = src
  const float* edge_w = (const float*)d_in[1];  // [E, F] float32
  float*       out    = (float*)d_out;          // [N, F] float32

  const long long E = (long long)in_sizes[0] / 2;
  const int       F = (int)((long long)in_sizes[1] / E);

  // 1) zero output
  {
    int n = out_size;
    int threads = 256;
    int blocks = (int)((n / 4 + threads - 1) / threads) + 1;  // +1 covers tail
    sna_zero_kernel<<<blocks, threads, 0, stream>>>(out, n);
  }

  // 2) scatter-add
  if (F == 48) {
    unsigned int total = (unsigned int)(E * 12);   // 19.2M work-items
    unsigned int threads = 256;
    unsigned int blocks = (total + threads - 1) / threads;
    sna_scatter_kernel<12><<<blocks, threads, 0, stream>>>(edge, edge_w, out, total);
  } else if ((F & 3) == 0) {
    // other multiples of 4: scalarized per-element path (still native atomics)
    long long total = E * (long long)F;
    unsigned int threads = 256;
    long long blocks = (total + threads - 1) / threads;
    sna_scatter_generic<<<(unsigned int)blocks, threads, 0, stream>>>(edge, edge_w, out, total, F);
  } else {
    long long total = E * (long long)F;
    unsigned int threads = 256;
    long long blocks = (total + threads - 1) / threads;
    sna_scatter_generic<<<(unsigned int)blocks, threads, 0, stream>>>(edge, edge_w, out, total, F);
  }
}